// DAALayer_90443421319697
// MI455X (gfx1250) — compile-verified
//
#include <hip/hip_runtime.h>
#include <cstdint>

#define IN_F   1024
#define OUT_F  512
#define BATCH  256

// ---- CDNA5 async global->LDS path (guarded so compile never breaks) ----
#if defined(__HIP_DEVICE_COMPILE__) && __has_builtin(__builtin_amdgcn_global_load_async_to_lds_b128)
#define USE_ASYNC_LDS 1
typedef int v4i __attribute__((vector_size(16)));
typedef __attribute__((address_space(1))) v4i* gv4i_p;   // global int4*
typedef __attribute__((address_space(3))) v4i* lv4i_p;   // LDS int4*
#else
#define USE_ASYNC_LDS 0
#endif

__device__ __forceinline__ float gumbelf(float u) {
    const float eps = 1e-10f;
    return -logf(-logf(u + eps) + eps);
}

// -----------------------------------------------------------------------------
// Kernel 1: build no-edge bitmask, stored TRANSPOSED: maskT[w * OUT_F + o]
// holds bits for inputs i = w*32 .. w*32+31 of output o (bit j -> i = w*32+j).
// One thread per (o, w) word: 2M logf total -- negligible.
// -----------------------------------------------------------------------------
__global__ void __launch_bounds__(256) daa_mask_kernel(
    const float* __restrict__ etc, const float* __restrict__ noise,
    uint32_t* __restrict__ maskT) {
  int p = blockIdx.x * 256 + threadIdx.x;   // 0 .. OUT_F*32-1
  int o = p >> 5;
  int w = p & 31;
  size_t base = ((size_t)o * IN_F + (size_t)w * 32) * 2;  // [O, I, 2] layout
  const float4* e4 = (const float4*)(etc + base);
  const float4* n4 = (const float4*)(noise + base);
  uint32_t word = 0;
  #pragma unroll 4
  for (int q = 0; q < 16; ++q) {
    float4 e = e4[q];
    float4 n = n4[q];
    // argmax ties -> index 0, so no_edge only when strictly greater
    if ((e.y + gumbelf(n.y)) > (e.x + gumbelf(n.x))) word |= (1u << (2 * q));
    if ((e.w + gumbelf(n.w)) > (e.z + gumbelf(n.z))) word |= (1u << (2 * q + 1));
  }
  maskT[(size_t)w * OUT_F + o] = word;
}

// -----------------------------------------------------------------------------
// Kernel 2: masked min/max reduction.
// Block = 4 batch rows x 128 outputs; thread = 1 output x 2 batch rows.
// Lane->output mapping keeps output parity uniform per wave32 (no min/max
// divergence). Mask tile is [w][o_local] in LDS: lane stride 2 words ->
// conflict-free; x reads are wave broadcasts.
// -----------------------------------------------------------------------------
template <bool IS_MIN>
__device__ __forceinline__ void masked_reduce(
    const float* __restrict__ xr0, const float* __restrict__ xr1,
    const uint32_t* __restrict__ sm, int o_local, float offs,
    float& a0, float& a1) {
  a0 = offs; a1 = offs;
  const float4* p0 = (const float4*)xr0;
  const float4* p1 = (const float4*)xr1;
  for (int w = 0; w < 32; ++w) {
    uint32_t m = sm[w * 128 + o_local];
    #pragma unroll
    for (int q = 0; q < 8; ++q) {
      float4 v0 = p0[w * 8 + q];
      float4 v1 = p1[w * 8 + q];
      uint32_t mm = m >> (4 * q);
      float e;
      e = (mm & 1u) ? offs : v0.x;  a0 = IS_MIN ? fminf(a0, e) : fmaxf(a0, e);
      e = (mm & 1u) ? offs : v1.x;  a1 = IS_MIN ? fminf(a1, e) : fmaxf(a1, e);
      e = (mm & 2u) ? offs : v0.y;  a0 = IS_MIN ? fminf(a0, e) : fmaxf(a0, e);
      e = (mm & 2u) ? offs : v1.y;  a1 = IS_MIN ? fminf(a1, e) : fmaxf(a1, e);
      e = (mm & 4u) ? offs : v0.z;  a0 = IS_MIN ? fminf(a0, e) : fmaxf(a0, e);
      e = (mm & 4u) ? offs : v1.z;  a1 = IS_MIN ? fminf(a1, e) : fmaxf(a1, e);
      e = (mm & 8u) ? offs : v0.w;  a0 = IS_MIN ? fminf(a0, e) : fmaxf(a0, e);
      e = (mm & 8u) ? offs : v1.w;  a1 = IS_MIN ? fminf(a1, e) : fmaxf(a1, e);
    }
  }
}

__global__ void __launch_bounds__(256) daa_reduce_kernel(
    const float* __restrict__ x, const uint32_t* __restrict__ maskT,
    float* __restrict__ out) {
  __shared__ __align__(16) float    sx[4 * IN_F];    // 16 KB: 4 batch rows
  __shared__ __align__(16) uint32_t sm[32 * 128];    // 16 KB: mask [w][o_local]
  const int o0 = blockIdx.x * 128;
  const int b0 = blockIdx.y * 4;
  const int t  = (int)threadIdx.x;

  // ---- stage tiles into LDS ----
#if USE_ASYNC_LDS
  {
    const char* gx = (const char*)(x + (size_t)b0 * IN_F);   // contiguous 16KB
    char* lx = (char*)sx;
    char* lm = (char*)sm;
    #pragma unroll
    for (int k = 0; k < 4; ++k) {
      int n = t + k * 256;                                   // 1024 b128 chunks
      __builtin_amdgcn_global_load_async_to_lds_b128(
          (gv4i_p)(gx + (size_t)n * 16), (lv4i_p)(lx + n * 16), 0, 0);
    }
    #pragma unroll
    for (int k = 0; k < 4; ++k) {
      int n = t + k * 256;
      int w = n >> 5, c = n & 31;                            // 512B row slices
      const char* gm = (const char*)(maskT + (size_t)w * OUT_F + o0 + c * 4);
      __builtin_amdgcn_global_load_async_to_lds_b128(
          (gv4i_p)gm, (lv4i_p)(lm + n * 16), 0, 0);
    }
  #if __has_builtin(__builtin_amdgcn_s_wait_asynccnt)
    __builtin_amdgcn_s_wait_asynccnt(0);
  #else
    asm volatile("s_wait_asynccnt 0" ::: "memory");
  #endif
  }
#else
  {
    const float4* gx4 = (const float4*)(x + (size_t)b0 * IN_F);
    float4* lx4 = (float4*)sx;
    uint4*  lm4 = (uint4*)sm;
    #pragma unroll
    for (int k = 0; k < 4; ++k) {
      int n = t + k * 256;
      lx4[n] = gx4[n];
      int w = n >> 5, c = n & 31;
      lm4[n] = *(const uint4*)(maskT + (size_t)w * OUT_F + o0 + c * 4);
    }
  }
#endif
  __syncthreads();

  const int bg = t >> 7;                                  // which pair of rows
  const int u  = t & 127;
  const int o_local = ((u >> 6) & 1) + ((u & 63) << 1);   // parity uniform/wave
  const int o = o0 + o_local;
  const float* xr0 = sx + (size_t)(2 * bg)     * IN_F;
  const float* xr1 = sx + (size_t)(2 * bg + 1) * IN_F;

  float a0, a1;
  if ((o & 1) == 0) masked_reduce<true >(xr0, xr1, sm, o_local,  2.0f, a0, a1);
  else              masked_reduce<false>(xr0, xr1, sm, o_local, -1.0f, a0, a1);

  out[(size_t)(b0 + 2 * bg)     * OUT_F + o] = a0;
  out[(size_t)(b0 + 2 * bg + 1) * OUT_F + o] = a1;
}

// -----------------------------------------------------------------------------
// Launcher. d_ws usage: 32 * OUT_F uint32 = 64 KB (transposed edge mask).
// -----------------------------------------------------------------------------
extern "C" void kernel_launch(void* const* d_in, const int* in_sizes, int n_in,
                              void* d_out, int out_size, void* d_ws, size_t ws_size,
                              hipStream_t stream) {
  const float* x     = (const float*)d_in[0];  // [BATCH, IN_F]
  const float* etc   = (const float*)d_in[1];  // [OUT_F, IN_F, 2]
  const float* noise = (const float*)d_in[2];  // [OUT_F, IN_F, 2]
  float* out = (float*)d_out;                  // [BATCH, OUT_F]
  uint32_t* maskT = (uint32_t*)d_ws;           // [32][OUT_F]

  daa_mask_kernel<<<dim3((OUT_F * 32) / 256), dim3(256), 0, stream>>>(
      etc, noise, maskT);
  daa_reduce_kernel<<<dim3(OUT_F / 128, BATCH / 4), dim3(256), 0, stream>>>(
      x, maskT, out);
}